// MultiHeadAttentionLayer_80942953660859
// MI455X (gfx1250) — compile-verified
//
#include <hip/hip_runtime.h>
#include <math.h>

typedef __attribute__((ext_vector_type(2))) float v2f;
typedef __attribute__((ext_vector_type(8))) float v8f;

#define N_NODES 50000
#define N_EDGES 800000
#define IN_DIM 96
#define HR 64                    // H*R
#define HD 96                    // H*D
#define N_ROW_TILES (N_NODES/16) // 3125
#define E_ROW_TILES (N_EDGES/16) // 50000
#define INV_SCALE 0.35355339059327373f  // 1/sqrt(8)

__device__ __forceinline__ v8f wmma4(v2f a, v2f b, v8f c) {
  // V_WMMA_F32_16X16X4_F32: D = A(16x4) x B(4x16) + C(16x16), fp32 exact
  return __builtin_amdgcn_wmma_f32_16x16x4_f32(
      /*neg_a=*/false, a, /*neg_b=*/false, b,
      /*c_mod=*/(short)0, c, /*reuse_a=*/false, /*reuse_b=*/false);
}

// ---------------------------------------------------------------- zero scratch
__global__ void zero_kernel(float* __restrict__ ws, int n) {
  int i = blockIdx.x * blockDim.x + threadIdx.x;
  int stride = gridDim.x * blockDim.x;
  for (; i < n; i += stride) ws[i] = 0.f;
}

// ---------------------------------------------------------------- column sums of h
__global__ void col_sum_kernel(const float* __restrict__ h, float* __restrict__ mean_sum) {
  int col = threadIdx.x;
  if (col >= IN_DIM) return;
  float acc = 0.f;
  for (int r = blockIdx.x; r < N_NODES; r += gridDim.x) acc += h[r * IN_DIM + col];
  atomicAdd(&mean_sum[col], acc);
}

// ---------------------------------------------------------------- qs = (mean_h @ q) / sqrt(R)
__global__ void qvec_kernel(const float* __restrict__ mean_sum, const float* __restrict__ q,
                            float* __restrict__ qs) {
  int c = threadIdx.x;
  if (c >= HR) return;
  float acc = 0.f;
  for (int i = 0; i < IN_DIM; ++i) acc += mean_sum[i] * q[i * HR + c];
  qs[c] = acc * (1.0f / (float)N_NODES) * INV_SCALE;
}

// ---------------------------------------------------------------- node projections
// Kq = (h @ p) * qs   (Qvec & 1/sqrt(R) folded in),  Vh = h @ V_w + V_b
// Weights staged TRANSPOSED in LDS: WT[col][k] so a B fragment is one b64 load.
#define NODE_WAVES 4
__global__ __launch_bounds__(NODE_WAVES * 32) void node_proj_kernel(
    const float* __restrict__ h, const float* __restrict__ p,
    const float* __restrict__ Vw, const float* __restrict__ Vb,
    const float* __restrict__ qs, float* __restrict__ Kq, float* __restrict__ Vh) {
  __shared__ float WT[160 * IN_DIM];  // [col][k]: cols 0..63 = p^T, 64..159 = V_w^T
  __shared__ float biasv[HD];
  __shared__ float qsl[HR];
  int tid = threadIdx.x;
  for (int i = tid; i < IN_DIM * HR; i += blockDim.x)
    WT[(i % HR) * IN_DIM + (i / HR)] = p[i];
  for (int i = tid; i < IN_DIM * HD; i += blockDim.x)
    WT[(64 + (i % HD)) * IN_DIM + (i / HD)] = Vw[i];
  for (int i = tid; i < HD; i += blockDim.x) biasv[i] = Vb[i];
  for (int i = tid; i < HR; i += blockDim.x) qsl[i] = qs[i];
  __syncthreads();

  int wave = tid >> 5, lane = tid & 31;
  int lrow = lane & 15;
  int koff = (lane < 16) ? 0 : 2;
  int mofs = (lane < 16) ? 0 : 8;
  int waveGlobal = blockIdx.x * NODE_WAVES + wave;
  int nWaves = gridDim.x * NODE_WAVES;

  for (int tile = waveGlobal; tile < N_ROW_TILES; tile += nWaves) {
    int rbase = tile * 16;
    const float* arow = h + (size_t)(rbase + lrow) * IN_DIM;

    for (int ct = 0; ct < 10; ++ct) {
      int col = ct * 16 + lrow;
      float binit = (ct >= 4) ? biasv[col - 64] : 0.f;
      v8f acc;
      for (int i = 0; i < 8; ++i) acc[i] = binit;
      const float* bcol = &WT[col * IN_DIM];
      for (int k = 0; k < IN_DIM; k += 4) {
        v2f a = *(const v2f*)(arow + k + koff);
        v2f b = *(const v2f*)(bcol + k + koff);   // single ds_load_b64
        acc = wmma4(a, b, acc);
      }
      if (ct < 4) {
        float qc = qsl[ct * 16 + lrow];
        for (int v = 0; v < 8; ++v)
          Kq[(size_t)(rbase + v + mofs) * HR + ct * 16 + lrow] = acc[v] * qc;
      } else {
        for (int v = 0; v < 8; ++v)
          Vh[(size_t)(rbase + v + mofs) * HD + (ct - 4) * 16 + lrow] = acc[v];
      }
    }
  }
}

// ---------------------------------------------------------------- fused edge pipeline
// per 16-edge tile: proj_e = e@pe_w + pe_b (WMMA) -> score = proj_e * Kq[src]
// -> s = exp(clip(head-sums)) -> e_out = score @ peo_w + peo_b (WMMA)
// -> atomic scatter of s*Vh[src] and s into wV/z
#define EDGE_WAVES 3
__global__ __launch_bounds__(EDGE_WAVES * 32) void edge_kernel(
    const float* __restrict__ e, const int* __restrict__ src, const int* __restrict__ dst,
    const float* __restrict__ Kq, const float* __restrict__ Vh,
    const float* __restrict__ pe_w, const float* __restrict__ pe_b,
    const float* __restrict__ peo_w, const float* __restrict__ peo_b,
    float* __restrict__ wV, float* __restrict__ z, float* __restrict__ e_out) {
  __shared__ float peT[HR * IN_DIM];          // pe_w^T:  [n][k]  64x96
  __shared__ float peoT[HD * HR];             // peo_w^T: [n][k]  96x64
  __shared__ float pebL[HR], peobL[HD];
  __shared__ float scoreLds[EDGE_WAVES][16 * HR];
  __shared__ float sbuf[EDGE_WAVES][16][8];
  __shared__ int   srcL[EDGE_WAVES][16], dstL[EDGE_WAVES][16];

  int tid = threadIdx.x;
  for (int i = tid; i < IN_DIM * HR; i += blockDim.x)
    peT[(i % HR) * IN_DIM + (i / HR)] = pe_w[i];
  for (int i = tid; i < HR * HD; i += blockDim.x)
    peoT[(i % HD) * HR + (i / HD)] = peo_w[i];
  for (int i = tid; i < HR; i += blockDim.x) pebL[i] = pe_b[i];
  for (int i = tid; i < HD; i += blockDim.x) peobL[i] = peo_b[i];
  __syncthreads();

  int wave = tid >> 5, lane = tid & 31;
  int lrow = lane & 15;
  int koff = (lane < 16) ? 0 : 2;
  int mofs = (lane < 16) ? 0 : 8;
  int waveGlobal = blockIdx.x * EDGE_WAVES + wave;
  int nWaves = gridDim.x * EDGE_WAVES;

  for (int tile = waveGlobal; tile < E_ROW_TILES; tile += nWaves) {
    int ebase = tile * 16;

    if (lane < 16) {
      int sv = src[ebase + lane], dv = dst[ebase + lane];
      srcL[wave][lane] = sv;
      dstL[wave][lane] = dv;
      __builtin_prefetch(&Kq[(size_t)sv * HR], 0, 0);  // global_prefetch_b8
      __builtin_prefetch(&Vh[(size_t)sv * HD], 0, 0);
    }
    __builtin_amdgcn_wave_barrier();

    // -- proj_e tile (4 col tiles of 16): accP = e_tile @ pe_w + pe_b
    const float* arow = e + (size_t)(ebase + lrow) * IN_DIM;
    v8f accP[4];
    for (int t = 0; t < 4; ++t) {
      float b0 = pebL[t * 16 + lrow];
      for (int i = 0; i < 8; ++i) accP[t][i] = b0;
    }
    for (int k = 0; k < IN_DIM; k += 4) {
      v2f a = *(const v2f*)(arow + k + koff);
      for (int t = 0; t < 4; ++t) {
        v2f b = *(const v2f*)(&peT[(t * 16 + lrow) * IN_DIM + k + koff]);
        accP[t] = wmma4(a, b, accP[t]);
      }
    }

    // -- score = proj_e * Kq[src]  (Kq already holds Qvec/sqrt(R) factor)
    for (int t = 0; t < 4; ++t) {
      int n = t * 16 + lrow;
      for (int v = 0; v < 8; ++v)
        accP[t][v] *= Kq[(size_t)srcL[wave][v + mofs] * HR + n];
    }

    // -- relayout score tile (D-layout -> row-major LDS, used as A operand below)
    for (int t = 0; t < 4; ++t)
      for (int v = 0; v < 8; ++v)
        scoreLds[wave][(v + mofs) * HR + t * 16 + lrow] = accP[t][v];

    // -- per-head row sums (R=8 lanes per head) -> s = exp(clip(sum,-5,5))
    for (int t = 0; t < 4; ++t) {
      for (int v = 0; v < 8; ++v) {
        float sum = accP[t][v];
        sum += __shfl_xor(sum, 1, 32);
        sum += __shfl_xor(sum, 2, 32);
        sum += __shfl_xor(sum, 4, 32);
        if ((lane & 7) == 0) {
          float sv = expf(fminf(fmaxf(sum, -5.f), 5.f));
          int hh = t * 2 + ((lane & 15) >> 3);
          sbuf[wave][v + mofs][hh] = sv;
        }
      }
    }
    __builtin_amdgcn_wave_barrier();

    // -- e_out = score @ peo_w + peo_b  (6 col tiles of 16)
    v8f accO[6];
    for (int t = 0; t < 6; ++t) {
      float b0 = peobL[t * 16 + lrow];
      for (int i = 0; i < 8; ++i) accO[t][i] = b0;
    }
    const float* sA = &scoreLds[wave][lrow * HR];
    for (int k = 0; k < HR; k += 4) {
      v2f a = *(const v2f*)(sA + k + koff);
      for (int t = 0; t < 6; ++t) {
        v2f b = *(const v2f*)(&peoT[(t * 16 + lrow) * HR + k + koff]);
        accO[t] = wmma4(a, b, accO[t]);
      }
    }
    for (int t = 0; t < 6; ++t)
      for (int v = 0; v < 8; ++v)
        e_out[(size_t)(ebase + v + mofs) * HD + t * 16 + lrow] = accO[t][v];

    // -- scatter aggregation (L2-resident accumulators)
    for (int idx = lane; idx < 16 * HD; idx += 32) {
      int m = idx / HD, c = idx % HD;
      float sv = sbuf[wave][m][c / 12];
      atomicAdd(&wV[(size_t)dstL[wave][m] * HD + c],
                sv * Vh[(size_t)srcL[wave][m] * HD + c]);
    }
    for (int idx = lane; idx < 16 * 8; idx += 32) {
      int m = idx >> 3, hh = idx & 7;
      atomicAdd(&z[(size_t)dstL[wave][m] * 8 + hh], sbuf[wave][m][hh]);
    }
    __builtin_amdgcn_wave_barrier();
  }
}

// ---------------------------------------------------------------- h_out = wV / (z + eps)
__global__ void finalize_kernel(const float* __restrict__ wV, const float* __restrict__ z,
                                float* __restrict__ h_out) {
  int i = blockIdx.x * blockDim.x + threadIdx.x;
  int stride = gridDim.x * blockDim.x;
  for (; i < N_NODES * HD; i += stride) {
    int n = i / HD, c = i % HD;
    h_out[i] = wV[i] / (z[n * 8 + c / 12] + 1e-6f);
  }
}

extern "C" void kernel_launch(void* const* d_in, const int* in_sizes, int n_in,
                              void* d_out, int out_size, void* d_ws, size_t ws_size,
                              hipStream_t stream) {
  const float* h    = (const float*)d_in[0];
  const float* e    = (const float*)d_in[1];
  const int*   src  = (const int*)d_in[2];
  const int*   dst  = (const int*)d_in[3];
  const float* p    = (const float*)d_in[4];
  const float* q    = (const float*)d_in[5];
  const float* Vw   = (const float*)d_in[6];
  const float* Vb   = (const float*)d_in[7];
  const float* pew  = (const float*)d_in[8];
  const float* peb  = (const float*)d_in[9];
  const float* peow = (const float*)d_in[10];
  const float* peob = (const float*)d_in[11];
  float* out = (float*)d_out;
  float* ws  = (float*)d_ws;

  // workspace layout (floats): mean_sum[96] qs[64] z[N*8] wV[N*96] Kq[N*64] Vh[N*96]
  float* mean_sum = ws;
  float* qs   = ws + 96;
  float* zacc = ws + 160;
  float* wV   = zacc + N_NODES * 8;
  float* Kq   = wV + N_NODES * HD;
  float* Vh   = Kq + N_NODES * HR;

  int zcount = 160 + N_NODES * 8 + N_NODES * HD;  // mean_sum + qs + z + wV
  zero_kernel<<<2048, 256, 0, stream>>>(ws, zcount);
  col_sum_kernel<<<256, 96, 0, stream>>>(h, mean_sum);
  qvec_kernel<<<1, 64, 0, stream>>>(mean_sum, q, qs);
  node_proj_kernel<<<256, NODE_WAVES * 32, 0, stream>>>(h, p, Vw, Vb, qs, Kq, Vh);
  edge_kernel<<<2000, EDGE_WAVES * 32, 0, stream>>>(
      e, src, dst, Kq, Vh, pew, peb, peow, peob, wV, zacc, out + (size_t)N_NODES * HD);
  finalize_kernel<<<4096, 256, 0, stream>>>(wV, zacc, out);
}